// SelectiveSSM_50328426775067
// MI455X (gfx1250) — compile-verified
//
#include <hip/hip_runtime.h>
#include <hip/hip_bf16.h>
#include <math.h>

// ---------------------------------------------------------------------------
// Selective SSM (Mamba block) forward for MI455X / gfx1250, wave32 + WMMA.
//   B=2, S=2048, D_MODEL=512, D_INNER=1024, D_STATE=64, D_CONV=4
// ---------------------------------------------------------------------------

#define SSM_D_MODEL 512
#define SSM_D_STATE 64
#define SSM_D_CONV  4
#define SSM_D_INNER 1024
#define SSM_BATCH   2
#define SSM_SEQ     2048
#define SSM_NTOK    (SSM_BATCH * SSM_SEQ)           // 4096 tokens
#define SSM_NPROJ   (2 * SSM_D_STATE + SSM_D_INNER) // 1152

#define SCAN_CH 64                     // chunks per sequence
#define SCAN_L  (SSM_SEQ / SCAN_CH)    // 32 timesteps per chunk

typedef __attribute__((ext_vector_type(2))) float v2f;
typedef __attribute__((ext_vector_type(8))) float v8f;

__device__ __forceinline__ float softplus_f(float x) {
    return fmaxf(x, 0.0f) + log1pf(__expf(-fabsf(x)));
}
__device__ __forceinline__ float silu_f(float x) {
    return x / (1.0f + __expf(-x));
}

// ---------------------------------------------------------------------------
// C[M,N] = X[M,K] @ W[N,K]^T + bias[N]   (row-major, ld == dim)
// f32 WMMA 16x16x4, LDS-staged, double-buffered (see round-2 notes).
// ---------------------------------------------------------------------------
__global__ __launch_bounds__(256) void wmma_gemm_xt(
    const float* __restrict__ X, const float* __restrict__ W,
    const float* __restrict__ bias, float* __restrict__ C,
    int M, int N, int K)
{
    constexpr int KC  = 32;
    constexpr int LDA = 36;                 // padded LDS row stride (floats)
    __shared__ float sA[2][128 * LDA];
    __shared__ float sB[2][64 * LDA];

    const int tid  = threadIdx.x;
    const int lane = tid & 31;
    const int wave = tid >> 5;
    const int m0   = blockIdx.y * 128;
    const int n0   = blockIdx.x * 64;
    const int half = lane >> 4;             // 0: K=0,1   1: K=2,3
    const int l15  = lane & 15;

    const int arow = tid >> 1;              // 0..127
    const int acol = (tid & 1) * 16;        // 4 float4s per thread
    const int brow = tid >> 2;              // 0..63
    const int bcol = (tid & 3) * 8;         // 2 float4s per thread

    const float* gA = X + (size_t)(m0 + arow) * K + acol;
    const float* gB = W + (size_t)(n0 + brow) * K + bcol;

    float4 ra[4], rb[2];
#pragma unroll
    for (int q = 0; q < 4; ++q) ra[q] = *(const float4*)(gA + q * 4);
#pragma unroll
    for (int q = 0; q < 2; ++q) rb[q] = *(const float4*)(gB + q * 4);

    v8f acc[4];
#pragma unroll
    for (int t = 0; t < 4; ++t) acc[t] = (v8f){};

    const int nch = K / KC;
    for (int c = 0; c < nch; ++c) {
        const int buf = c & 1;
#pragma unroll
        for (int q = 0; q < 4; ++q)
            *(float4*)&sA[buf][arow * LDA + acol + q * 4] = ra[q];
#pragma unroll
        for (int q = 0; q < 2; ++q)
            *(float4*)&sB[buf][brow * LDA + bcol + q * 4] = rb[q];
        __syncthreads();

        if (c + 1 < nch) {
            const float* gA2 = gA + (c + 1) * KC;
            const float* gB2 = gB + (c + 1) * KC;
#pragma unroll
            for (int q = 0; q < 4; ++q) ra[q] = *(const float4*)(gA2 + q * 4);
#pragma unroll
            for (int q = 0; q < 2; ++q) rb[q] = *(const float4*)(gB2 + q * 4);
        }

        const float* lA = &sA[buf][(wave * 16 + l15) * LDA + half * 2];
        const float* lB = &sB[buf][l15 * LDA + half * 2];
#pragma unroll
        for (int kk = 0; kk < 8; ++kk) {
            v2f a = *(const v2f*)(lA + kk * 4);
#pragma unroll
            for (int t = 0; t < 4; ++t) {
                v2f b = *(const v2f*)(lB + t * 16 * LDA + kk * 4);
                acc[t] = __builtin_amdgcn_wmma_f32_16x16x4_f32(
                    false, a, false, b, (short)0, acc[t], false, false);
            }
        }
        __syncthreads();
    }

    const int mrow = m0 + wave * 16 + half * 8;
#pragma unroll
    for (int t = 0; t < 4; ++t) {
        const int col = n0 + t * 16 + l15;
        const float bv = bias[col];
#pragma unroll
        for (int r = 0; r < 8; ++r)
            C[(size_t)(mrow + r) * N + col] = acc[t][r] + bv;
    }
}

// ---------------------------------------------------------------------------
// Depthwise causal conv1d (K=4) + bias + SiLU.
// ---------------------------------------------------------------------------
__global__ __launch_bounds__(256) void conv_silu_kernel(
    const float* __restrict__ xr, const float* __restrict__ cw,
    const float* __restrict__ cb, float* __restrict__ xc)
{
    const int idx = blockIdx.x * blockDim.x + threadIdx.x;
    if (idx >= SSM_NTOK * SSM_D_INNER) return;
    const int d = idx & (SSM_D_INNER - 1);
    const int t = (idx >> 10) & (SSM_SEQ - 1);
    const int b = idx >> 21;

    float acc = cb[d];
#pragma unroll
    for (int j = 0; j < SSM_D_CONV; ++j) {
        const int tt = t - (SSM_D_CONV - 1) + j;
        if (tt >= 0)
            acc = fmaf(cw[d * SSM_D_CONV + j],
                       xr[(size_t)(b * SSM_SEQ + tt) * (2 * SSM_D_INNER) + d],
                       acc);
    }
    xc[(size_t)(b * SSM_SEQ + t) * SSM_D_INNER + d] = silu_f(acc);
}

// ---------------------------------------------------------------------------
// Chunked parallel scan, pass 1: local chunk scan from h=0.
// grid = B * CH * (D_INNER/256) = 512 blocks.  blockIdx: [b | c | dblk]
// Emits per (b,c,d): chunk-end state hEnd[64] and sum of deltas (the chunk
// decay product is exp(A_s * sum_delta) -- one scalar carries all 64 states).
// ---------------------------------------------------------------------------
__global__ __launch_bounds__(256) void scan_partial_kernel(
    const float* __restrict__ ssm, const float* __restrict__ xc,
    const float* __restrict__ A_log,
    float* __restrict__ hEnd, float* __restrict__ sd)
{
    __shared__ float sB[SSM_D_STATE];
    const int tid  = threadIdx.x;
    const int dblk = blockIdx.x & 3;
    const int c    = (blockIdx.x >> 2) & (SCAN_CH - 1);
    const int b    = blockIdx.x >> 8;
    const int d    = dblk * 256 + tid;

    float An[SSM_D_STATE], h[SSM_D_STATE];
#pragma unroll
    for (int s = 0; s < SSM_D_STATE; ++s) {
        An[s] = -__expf(A_log[(size_t)d * SSM_D_STATE + s]);
        h[s]  = 0.0f;
    }

    float sumd = 0.0f;
    for (int tt = 0; tt < SCAN_L; ++tt) {
        const size_t row = (size_t)(b * SSM_SEQ + c * SCAN_L + tt);
        const float* srow = ssm + row * SSM_NPROJ;

        __syncthreads();
        if (tid < SSM_D_STATE) sB[tid] = srow[tid];
        __syncthreads();

        const float delta = softplus_f(srow[2 * SSM_D_STATE + d]);
        const float dx = delta * xc[row * SSM_D_INNER + d];
        sumd += delta;
#pragma unroll
        for (int s = 0; s < SSM_D_STATE; ++s)
            h[s] = fmaf(__expf(delta * An[s]), h[s], dx * sB[s]);
    }

    float* hp = hEnd + (((size_t)(b * SCAN_CH + c) * SSM_D_INNER + d) << 6);
#pragma unroll
    for (int s = 0; s < SSM_D_STATE; s += 4)
        *(float4*)(hp + s) = make_float4(h[s], h[s + 1], h[s + 2], h[s + 3]);
    sd[(b * SCAN_CH + c) * SSM_D_INNER + d] = sumd;
}

// ---------------------------------------------------------------------------
// Pass 2: sequential combine across the 64 chunks (tiny), one thread per
// (b,d,s) = 131072 threads.  Writes each chunk's initial state h0.
// ---------------------------------------------------------------------------
__global__ __launch_bounds__(256) void scan_combine_kernel(
    const float* __restrict__ A_log, const float* __restrict__ sd,
    const float* __restrict__ hEnd, float* __restrict__ h0)
{
    const int gid = blockIdx.x * 256 + threadIdx.x;  // (b*Di + d)*64 + s
    const int s  = gid & (SSM_D_STATE - 1);
    const int bd = gid >> 6;
    const int d  = bd & (SSM_D_INNER - 1);
    const int b  = bd >> 10;

    const float An = -__expf(A_log[(size_t)d * SSM_D_STATE + s]);
    float H = 0.0f;
    for (int c = 0; c < SCAN_CH; ++c) {
        const int    bcd = (b * SCAN_CH + c) * SSM_D_INNER + d;
        const size_t off = ((size_t)bcd << 6) + s;
        h0[off] = H;
        H = fmaf(__expf(An * sd[bcd]), H, hEnd[off]);
    }
}

// ---------------------------------------------------------------------------
// Pass 3: re-run each chunk seeded with h0, emit y_t (+ fused SiLU gate).
// Same block layout as pass 1.
// ---------------------------------------------------------------------------
__global__ __launch_bounds__(256) void scan_final_kernel(
    const float* __restrict__ ssm, const float* __restrict__ xc,
    const float* __restrict__ xr, const float* __restrict__ A_log,
    const float* __restrict__ Dvec, const float* __restrict__ h0,
    float* __restrict__ yg)
{
    __shared__ float sBC[2 * SSM_D_STATE];
    const int tid  = threadIdx.x;
    const int dblk = blockIdx.x & 3;
    const int c    = (blockIdx.x >> 2) & (SCAN_CH - 1);
    const int b    = blockIdx.x >> 8;
    const int d    = dblk * 256 + tid;

    float An[SSM_D_STATE], h[SSM_D_STATE];
#pragma unroll
    for (int s = 0; s < SSM_D_STATE; ++s)
        An[s] = -__expf(A_log[(size_t)d * SSM_D_STATE + s]);

    const float* hp = h0 + (((size_t)(b * SCAN_CH + c) * SSM_D_INNER + d) << 6);
#pragma unroll
    for (int s = 0; s < SSM_D_STATE; s += 4) {
        const float4 v = *(const float4*)(hp + s);
        h[s] = v.x; h[s + 1] = v.y; h[s + 2] = v.z; h[s + 3] = v.w;
    }
    const float Dd = Dvec[d];

    for (int tt = 0; tt < SCAN_L; ++tt) {
        const size_t row = (size_t)(b * SSM_SEQ + c * SCAN_L + tt);
        const float* srow = ssm + row * SSM_NPROJ;

        __syncthreads();
        if (tid < 2 * SSM_D_STATE) sBC[tid] = srow[tid];
        __syncthreads();

        const float delta = softplus_f(srow[2 * SSM_D_STATE + d]);
        const float xt = xc[row * SSM_D_INNER + d];
        const float dx = delta * xt;

        float y = 0.0f;
#pragma unroll
        for (int s = 0; s < SSM_D_STATE; ++s) {
            h[s] = fmaf(__expf(delta * An[s]), h[s], dx * sBC[s]);
            y = fmaf(h[s], sBC[SSM_D_STATE + s], y);
        }
        y = fmaf(Dd, xt, y);

        const float g = silu_f(xr[row * (2 * SSM_D_INNER) + SSM_D_INNER + d]);
        yg[row * SSM_D_INNER + d] = y * g;
    }
}

// ---------------------------------------------------------------------------
// Launch
// ---------------------------------------------------------------------------
extern "C" void kernel_launch(void* const* d_in, const int* in_sizes, int n_in,
                              void* d_out, int out_size, void* d_ws, size_t ws_size,
                              hipStream_t stream) {
    const float* x      = (const float*)d_in[0];
    const float* in_w   = (const float*)d_in[1];   // (2048, 512)
    const float* in_b   = (const float*)d_in[2];
    const float* conv_w = (const float*)d_in[3];   // (1024, 1, 4)
    const float* conv_b = (const float*)d_in[4];
    const float* xp_w   = (const float*)d_in[5];   // (1152, 1024)
    const float* xp_b   = (const float*)d_in[6];
    const float* A_log  = (const float*)d_in[7];   // (1024, 64)
    const float* Dvec   = (const float*)d_in[8];
    const float* out_w  = (const float*)d_in[9];   // (512, 1024)
    const float* out_b  = (const float*)d_in[10];
    float* out = (float*)d_out;

    char* ws = (char*)d_ws;
    float* xr   = (float*)(ws + 0);           // 4096 x 2048   (33.5 MB)
    float* xc   = (float*)(ws + 33554432);    // 4096 x 1024   (16.8 MB)
    float* ssm  = (float*)(ws + 50331648);    // 4096 x 1152   (18.9 MB)
    float* yg   = (float*)(ws + 69206016);    // 4096 x 1024   (16.8 MB)
    float* hEnd = (float*)(ws + 85983232);    // 2*64*1024*64  (33.5 MB)
    float* h0   = (float*)(ws + 119537664);   // 2*64*1024*64  (33.5 MB)
    float* sd   = (float*)(ws + 153092096);   // 2*64*1024     (0.5 MB)

    dim3 blk(256);

    // 1) in_proj: (4096,512) @ (2048,512)^T -> (4096,2048)
    wmma_gemm_xt<<<dim3(2048 / 64, SSM_NTOK / 128), blk, 0, stream>>>(
        x, in_w, in_b, xr, SSM_NTOK, 2 * SSM_D_INNER, SSM_D_MODEL);

    // 2) depthwise conv + SiLU
    conv_silu_kernel<<<(SSM_NTOK * SSM_D_INNER) / 256, blk, 0, stream>>>(
        xr, conv_w, conv_b, xc);

    // 3) x_proj: (4096,1024) @ (1152,1024)^T -> (4096,1152)
    wmma_gemm_xt<<<dim3(SSM_NPROJ / 64, SSM_NTOK / 128), blk, 0, stream>>>(
        xc, xp_w, xp_b, ssm, SSM_NTOK, SSM_NPROJ, SSM_D_INNER);

    // 4) chunked selective scan (3 passes) + fused SiLU gate
    const int scan_blocks = SSM_BATCH * SCAN_CH * (SSM_D_INNER / 256); // 512
    scan_partial_kernel<<<dim3(scan_blocks), blk, 0, stream>>>(
        ssm, xc, A_log, hEnd, sd);
    scan_combine_kernel<<<dim3((SSM_BATCH * SSM_D_INNER * SSM_D_STATE) / 256),
                         blk, 0, stream>>>(A_log, sd, hEnd, h0);
    scan_final_kernel<<<dim3(scan_blocks), blk, 0, stream>>>(
        ssm, xc, xr, A_log, Dvec, h0, yg);

    // 5) out_proj: (4096,1024) @ (512,1024)^T -> (4096,512)
    wmma_gemm_xt<<<dim3(SSM_D_MODEL / 64, SSM_NTOK / 128), blk, 0, stream>>>(
        yg, out_w, out_b, out, SSM_NTOK, SSM_D_MODEL, SSM_D_INNER);
}